// RoIAlign_9698036154943
// MI455X (gfx1250) — compile-verified
//
#include <hip/hip_runtime.h>

// ---------------- problem constants (from reference setup_inputs) ------------
#define BB   2      // batches
#define CC   256    // channels
#define HH   200    // feature height
#define WW   304    // feature width
#define NR   1000   // rois per batch
#define RR   (BB*NR)
#define PP   7      // pooled size
// sampling ratio = 2 -> 14 samples/axis -> 28 taps/axis

typedef float v2f __attribute__((ext_vector_type(2)));
typedef float v8f __attribute__((ext_vector_type(8)));

// ---------------------------------------------------------------------------
// Kernel 1: (B,C,H,W) -> (B,H,W,C) channels-last transpose, tiled via LDS.
// Makes every bilinear tap a 64B-contiguous 16-channel read (WMMA A striping).
// ---------------------------------------------------------------------------
__global__ __launch_bounds__(256) void fm_transpose(const float* __restrict__ fm,
                                                    float* __restrict__ fmT) {
  __shared__ float tile[32][33];          // +1 pad: no LDS bank conflicts
  const int HWl = HH * WW;                // 60800 = 1900*32 (exact tiles)
  const int tx = threadIdx.x, ty = threadIdx.y;
  const int b  = blockIdx.z;
  const int n0 = blockIdx.x * 32;         // along H*W
  const int m0 = blockIdx.y * 32;         // along C (256 = 8*32, exact)
#pragma unroll
  for (int i = 0; i < 4; ++i) {
    int m = m0 + ty + i * 8;
    tile[ty + i * 8][tx] = fm[((size_t)b * CC + m) * HWl + (n0 + tx)];
  }
  __syncthreads();
#pragma unroll
  for (int i = 0; i < 4; ++i) {
    int n = n0 + ty + i * 8;
    fmT[((size_t)b * HWl + n) * CC + (m0 + tx)] = tile[tx][ty + i * 8];
  }
}

// ---------------------------------------------------------------------------
// Kernel 2: RoIAlign via V_WMMA_F32_16X16X4_F32.
//   grid = (2000, 4), block = 128 (4 wave32s). Wave w of block.y handles
//   channel tile c0 = (blockIdx.y*4 + w)*16 of one ROI.
//
//   A (16x4 f32)  : lanes 0-15 = 16 channels, K = 4 y-taps (VGPR0: t0/t2 per
//                   lane half, VGPR1: t1/t3) -> direct gathered loads.
//   B (4x16 f32)  : lane n = bin-row ph; B_k nonzero only at ph==k, value
//                   0.25*wy(t)*wx(u)  (0.5*0.5 = the 2x2 sample mean).
//   D (16x16 f32) : column ph = out[c, ph, pw]; one accumulator per pw chains
//                   all 4 x-taps * 7 K-steps (28 WMMAs) -> stored directly.
// ---------------------------------------------------------------------------
__global__ __launch_bounds__(128) void roialign_wmma(const float* __restrict__ feat,
                                                     const float* __restrict__ props,
                                                     float* __restrict__ out,
                                                     int yStride, int xStride,
                                                     int cStride) {
  const int r    = blockIdx.x;
  const int wave = threadIdx.x >> 5;      // gfx1250 is wave32
  const int lane = threadIdx.x & 31;
  const int half = lane >> 4;             // 0: K taps t0/t1, 1: K taps t2/t3
  const int lm   = lane & 15;             // channel-in-tile / ph column
  const int c0   = (blockIdx.y * 4 + wave) * 16;
  const int batch = r / NR;

  const float4 p = ((const float4*)props)[r];
  const float x1 = p.x * 0.25f, y1 = p.y * 0.25f;
  const float x2 = p.z * 0.25f, y2 = p.w * 0.25f;
  const float bin_w = fmaxf(x2 - x1, 1.0f) * (1.0f / 7.0f);
  const float bin_h = fmaxf(y2 - y1, 1.0f) * (1.0f / 7.0f);

  // Per-lane y-tap row offsets + y-weights (t = 4k + j + 2*half, ph(t)=k).
  int   rowOff[7][2];
  float ywB[7][2];
#pragma unroll
  for (int k = 0; k < 7; ++k) {
#pragma unroll
    for (int j = 0; j < 2; ++j) {
      const int t    = 4 * k + j + 2 * half;
      const int sy   = t >> 1;
      const int isHi = t & 1;
      const float coord = y1 + ((float)sy + 0.5f) * 0.5f * bin_h;
      const bool  valid = (coord >= -1.0f) && (coord <= (float)HH);
      const float cc = fminf(fmaxf(coord, 0.0f), (float)(HH - 1));
      const int   lo = (int)floorf(cc);
      const int   hi = min(lo + 1, HH - 1);
      const float fy = cc - (float)lo;
      rowOff[k][j] = (isHi ? hi : lo) * yStride;
      float w = isHi ? fy : (1.0f - fy);
      w = valid ? w : 0.0f;
      ywB[k][j] = (lm == k) ? (0.5f * w) : 0.0f;   // B nonzero only at ph==k
    }
  }

  const float* fb = feat + (size_t)batch * HH * WW * CC;

#pragma unroll
  for (int pw = 0; pw < 7; ++pw) {
    v8f acc = {0.0f, 0.0f, 0.0f, 0.0f, 0.0f, 0.0f, 0.0f, 0.0f};
#pragma unroll
    for (int q = 0; q < 4; ++q) {                 // 4 x-taps of this pw
      const int u    = 4 * pw + q;
      const int sx   = u >> 1;
      const int isHi = u & 1;
      const float coordx = x1 + ((float)sx + 0.5f) * 0.5f * bin_w;
      const bool  validx = (coordx >= -1.0f) && (coordx <= (float)WW);
      const float cx  = fminf(fmaxf(coordx, 0.0f), (float)(WW - 1));
      const int   xlo = (int)floorf(cx);
      const int   xhi = min(xlo + 1, WW - 1);
      const float fx  = cx - (float)xlo;
      const int   xcol = isHi ? xhi : xlo;
      float wx = isHi ? fx : (1.0f - fx);
      wx = validx ? wx : 0.0f;
      const float xw = 0.5f * wx;                 // folded into B per WMMA
      const int colOff = xcol * xStride + (c0 + lm) * cStride;
#pragma unroll
      for (int k = 0; k < 7; ++k) {               // K = 28 y-taps, 4 per step
        v2f a;
        a.x = fb[rowOff[k][0] + colOff];
        a.y = fb[rowOff[k][1] + colOff];
        v2f b;
        b.x = ywB[k][0] * xw;
        b.y = ywB[k][1] * xw;
        acc = __builtin_amdgcn_wmma_f32_16x16x4_f32(
            /*neg_a=*/false, a, /*neg_b=*/false, b,
            /*c_mod=*/(short)0, acc, /*reuse_a=*/false, /*reuse_b=*/false);
      }
    }
    // acc[v] = out[c0 + v + 8*half, ph=lm, pw]; columns ph>=7 are zero.
    if (lm < 7) {
      const size_t obase =
          (((size_t)r * CC + (c0 + 8 * half)) * PP + lm) * PP + pw;
#pragma unroll
      for (int v = 0; v < 8; ++v) out[obase + (size_t)v * (PP * PP)] = acc[v];
    }
  }
}

// ---------------------------------------------------------------------------
extern "C" void kernel_launch(void* const* d_in, const int* in_sizes, int n_in,
                              void* d_out, int out_size, void* d_ws, size_t ws_size,
                              hipStream_t stream) {
  const float* fm    = (const float*)d_in[0];   // (2,256,200,304) f32
  const float* props = (const float*)d_in[1];   // (2,1000,4) f32
  float* out = (float*)d_out;                   // (2000,256,7,7) f32

  const size_t needT = (size_t)BB * HH * WW * CC * sizeof(float); // 124.5 MB
  if (ws_size >= needT) {
    float* fmT = (float*)d_ws;
    dim3 tb(32, 8, 1), tg((HH * WW) / 32, CC / 32, BB);
    fm_transpose<<<tg, tb, 0, stream>>>(fm, fmT);
    dim3 mb(128, 1, 1), mg(RR, CC / 64, 1);
    // channels-last strides: y->W*C, x->C, c->1  (coalesced 64B A-gathers)
    roialign_wmma<<<mg, mb, 0, stream>>>(fmT, props, out, WW * CC, CC, 1);
  } else {
    // Fallback: same math straight from channels-first layout (correct, slower)
    dim3 mb(128, 1, 1), mg(RR, CC / 64, 1);
    roialign_wmma<<<mg, mb, 0, stream>>>(fm, props, out, WW, 1, HH * WW);
  }
}